// CombinedLoss_59674275610971
// MI455X (gfx1250) — compile-verified
//
#include <hip/hip_runtime.h>
#include <hip/hip_bf16.h>
#include <math.h>

#define EDT_CA 0.955f
#define EDT_CB 1.3693f
#define BIGF   1000000.0f
#define IMGS   16
#define HH     1024
#define WW     1024

typedef __attribute__((ext_vector_type(2))) float v2f;
typedef __attribute__((ext_vector_type(8))) float v8f;

#if __has_builtin(__builtin_amdgcn_wmma_f32_16x16x4_f32)
#define HAVE_WMMA_F32X4 1
#endif

#if defined(__gfx1250__) && \
    __has_builtin(__builtin_amdgcn_global_load_async_to_lds_b32) && \
    __has_builtin(__builtin_amdgcn_s_wait_asynccnt)
#define EDT_ASYNC 1
#endif

// ---------- async copy helpers (CDNA5 async-to-LDS path, guarded) ----------
__device__ __forceinline__ void async_cp_b32(const void* g, void* l) {
#if defined(EDT_ASYNC)
  int* gp = (int*)g;  // drop const (builtin takes non-const)
  int* lp = (int*)l;
  __builtin_amdgcn_global_load_async_to_lds_b32(
      (__attribute__((address_space(1))) int*)gp,
      (__attribute__((address_space(3))) int*)lp,
      0, 0);
#else
  *(int*)l = *(const int*)g;
#endif
}
__device__ __forceinline__ void wait_async() {
#if defined(EDT_ASYNC)
  __builtin_amdgcn_s_wait_asynccnt(0);
#endif
}

// ---------- wave32 sum via V_WMMA_F32_16X16X4_F32 (ones-matrix trick) ----------
// A vgpr0 = p, vgpr1 = 0  ->  A[m][0]=p_m, A[m][2]=p_{m+16}
// B = ones                ->  D[m][n] = p_m + p_{m+16}
// per-lane sum of the 8 C/D vgprs: lane0 = sum_{m=0..7}, lane16 = sum_{m=8..15}
__device__ __forceinline__ float wave_sum(float p) {
#if defined(HAVE_WMMA_F32X4)
  v2f a; a[0] = p;   a[1] = 0.0f;
  v2f b; b[0] = 1.0f; b[1] = 1.0f;
  v8f c = {0.f, 0.f, 0.f, 0.f, 0.f, 0.f, 0.f, 0.f};
  c = __builtin_amdgcn_wmma_f32_16x16x4_f32(false, a, false, b, (short)0, c,
                                            false, false);
  float t = ((c[0] + c[1]) + (c[2] + c[3])) + ((c[4] + c[5]) + (c[6] + c[7]));
  return __shfl(t, 0, 32) + __shfl(t, 16, 32);
#else
  #pragma unroll
  for (int d = 16; d; d >>= 1) p += __shfl_xor(p, d, 32);
  return p;
#endif
}

// ---------- block-wide sum (all threads active; nwaves = blockDim/32) ----------
__device__ __forceinline__ float block_sum(float v, float* red, int nwaves) {
  const int lane = threadIdx.x & 31;
  const int wave = threadIdx.x >> 5;
  float ws = wave_sum(v);
  if (lane == 0) red[wave] = ws;
  __syncthreads();
  if (wave == 0) {
    float r = (lane < nwaves) ? red[lane] : 0.0f;
    float t = wave_sum(r);
    if (lane == 0) red[0] = t;
  }
  __syncthreads();
  float total = red[0];
  __syncthreads();
  return total;
}

// ---------- block-wide inclusive min-scan, left->right (1024 threads) ----------
__device__ __forceinline__ float block_minscan_fwd(float s, float* waveAgg,
                                                   int lane, int wave) {
  #pragma unroll
  for (int d = 1; d < 32; d <<= 1) {
    float t = __shfl_up(s, d, 32);
    if (lane >= d) s = fminf(s, t);
  }
  if (lane == 31) waveAgg[wave] = s;
  __syncthreads();
  if (wave == 0) {
    float t = waveAgg[lane];
    #pragma unroll
    for (int d = 1; d < 32; d <<= 1) {
      float u = __shfl_up(t, d, 32);
      if (lane >= d) t = fminf(t, u);
    }
    waveAgg[lane] = t;
  }
  __syncthreads();
  if (wave > 0) s = fminf(s, waveAgg[wave - 1]);
  return s;
}

// ---------- block-wide inclusive min-scan, right->left (1024 threads) ----------
__device__ __forceinline__ float block_minscan_rev(float s, float* waveAgg,
                                                   int lane, int wave) {
  #pragma unroll
  for (int d = 1; d < 32; d <<= 1) {
    float t = __shfl_down(s, d, 32);
    if (lane + d < 32) s = fminf(s, t);
  }
  if (lane == 0) waveAgg[wave] = s;
  __syncthreads();
  if (wave == 0) {
    float t = waveAgg[lane];
    #pragma unroll
    for (int d = 1; d < 32; d <<= 1) {
      float u = __shfl_down(t, d, 32);
      if (lane + d < 32) t = fminf(t, u);
    }
    waveAgg[lane] = t;
  }
  __syncthreads();
  if (wave < 31) s = fminf(s, waveAgg[wave + 1]);
  return s;
}

// ============================ kernels ============================

__global__ void init_acc_kernel(float* __restrict__ acc) {
  if (threadIdx.x < 128) acc[threadIdx.x] = 0.0f;
}

// Focal + dice sums. grid = (256, IMGS), block = 256.
// acc[0]       : focal numerator (global sum of w*bce)
// acc[1+i]     : sum(sig*tf)   per image
// acc[17+i]    : sum(sig)      per image
// acc[33+i]    : sum(tf)       per image
__global__ __launch_bounds__(256) void focal_dice_kernel(
    const float* __restrict__ pred, const int* __restrict__ target,
    float* __restrict__ acc) {
  __shared__ float red[32];
  const int img = blockIdx.y;
  const size_t base =
      (size_t)img * (HH * WW) + (size_t)blockIdx.x * 4096;
  float fsum = 0.f, ssum = 0.f, tsum = 0.f, stsum = 0.f;
  #pragma unroll 4
  for (int u = 0; u < 16; ++u) {
    size_t idx = base + (size_t)u * 256 + threadIdx.x;
    float x = pred[idx];
    int tg = target[idx];
    float tf = (float)tg;
    float s = 1.f / (1.f + expf(-x));
    float bce = fmaxf(x, 0.f) + log1pf(expf(-fabsf(x))) - x * tf;
    float pt = tg ? s : (1.f - s);
    float alpha = tg ? 0.25f : 0.75f;
    float om = 1.f - pt;
    fsum += alpha * om * om * bce;
    ssum += s;
    tsum += tf;
    stsum += s * tf;
  }
  float t;
  t = block_sum(fsum, red, 8);
  if (threadIdx.x == 0) atomicAdd(&acc[0], t);
  t = block_sum(stsum, red, 8);
  if (threadIdx.x == 0) atomicAdd(&acc[1 + img], t);
  t = block_sum(ssum, red, 8);
  if (threadIdx.x == 0) atomicAdd(&acc[17 + img], t);
  t = block_sum(tsum, red, 8);
  if (threadIdx.x == 0) atomicAdd(&acc[33 + img], t);
}

// Chamfer EDT: one block per (image, polarity). grid = 32, block = 1024.
// pol 0: mask = (t>0)  -> "pos" field, weighted by (1-s)
// pol 1: mask = !(t>0) -> "neg" field, weighted by s
// acc[49+bx] : sum(weight * field)   (unnormalized; divide by max later)
// acc[81+bx] : max(field)
__global__ __launch_bounds__(1024) void edt_kernel(
    const int* __restrict__ target, const float* __restrict__ pred,
    float* __restrict__ acc, float* __restrict__ dfbase) {
  __shared__ float prevRow[WW + 2];
  __shared__ float waveAgg[32];
  __shared__ float red[32];
  __shared__ int   trow[2][WW];
  __shared__ float frow[2][WW];
  __shared__ float prow[2][WW];

  const int j = threadIdx.x;
  const int lane = j & 31;
  const int wave = j >> 5;
  const int bx = blockIdx.x;
  const int img = bx >> 1;
  const int pol = bx & 1;
  const size_t gbase = (size_t)img * (HH * WW);
  float* Df = dfbase + (size_t)bx * (HH * WW);
  const float aj = EDT_CA * (float)j;

  if (j == 0) { prevRow[0] = BIGF; prevRow[WW + 1] = BIGF; }
  prevRow[j + 1] = BIGF;

  // ---- forward pass (top -> bottom, left->right cummin) ----
  async_cp_b32(&target[gbase + j], &trow[0][j]);
  wait_async();
  __syncthreads();

  for (int i = 0; i < HH; ++i) {
    int tg = trow[i & 1][j];
    if (i + 1 < HH)
      async_cp_b32(&target[gbase + (size_t)(i + 1) * WW + j],
                   &trow[(i + 1) & 1][j]);
    float cur = (pol == 0) ? ((tg > 0) ? BIGF : 0.f)
                           : ((tg > 0) ? 0.f : BIGF);
    float pc = prevRow[j + 1];
    float pl = prevRow[j + 2];
    float pr = prevRow[j];
    float cand = fminf(fminf(cur, pc + EDT_CA), fminf(pl, pr) + EDT_CB);
    float v = block_minscan_fwd(cand - aj, waveAgg, lane, wave);
    float d = v + aj;
    Df[(size_t)i * WW + j] = d;
    prevRow[j + 1] = d;
    wait_async();
    __syncthreads();
  }
  __threadfence_block();
  prevRow[j + 1] = BIGF;
  __syncthreads();

  // ---- backward pass (bottom -> top, right->left cummin), fused with
  //      per-pixel sigmoid weighting, running sum and running max ----
  float sumAcc = 0.f;
  float maxAcc = 0.f;  // distances are >= 0
  async_cp_b32(&Df[(size_t)(HH - 1) * WW + j], &frow[(HH - 1) & 1][j]);
  async_cp_b32(&pred[gbase + (size_t)(HH - 1) * WW + j], &prow[(HH - 1) & 1][j]);
  wait_async();
  __syncthreads();

  for (int i = HH - 1; i >= 0; --i) {
    float cur = frow[i & 1][j];
    float x   = prow[i & 1][j];
    if (i - 1 >= 0) {
      async_cp_b32(&Df[(size_t)(i - 1) * WW + j], &frow[(i - 1) & 1][j]);
      async_cp_b32(&pred[gbase + (size_t)(i - 1) * WW + j],
                   &prow[(i - 1) & 1][j]);
    }
    float pc = prevRow[j + 1];
    float pl = prevRow[j + 2];
    float pr = prevRow[j];
    float cand = fminf(fminf(cur, pc + EDT_CA), fminf(pl, pr) + EDT_CB);
    float v = block_minscan_rev(cand + aj, waveAgg, lane, wave);
    float d = v - aj;
    float s = 1.f / (1.f + expf(-x));
    float wgt = pol ? s : (1.f - s);
    sumAcc += wgt * d;
    maxAcc = fmaxf(maxAcc, d);
    prevRow[j + 1] = d;
    wait_async();
    __syncthreads();
  }

  float tot = block_sum(sumAcc, red, 32);
  #pragma unroll
  for (int dlt = 16; dlt; dlt >>= 1)
    maxAcc = fmaxf(maxAcc, __shfl_xor(maxAcc, dlt, 32));
  if (lane == 0) red[wave] = maxAcc;
  __syncthreads();
  if (wave == 0) {
    float m = red[lane];
    #pragma unroll
    for (int dlt = 16; dlt; dlt >>= 1) m = fmaxf(m, __shfl_xor(m, dlt, 32));
    if (lane == 0) {
      acc[49 + bx] = tot;
      acc[81 + bx] = m;
    }
  }
}

__global__ void finalize_kernel(const float* __restrict__ acc,
                                float* __restrict__ out) {
  if (threadIdx.x != 0 || blockIdx.x != 0) return;
  const float N = (float)IMGS * (float)HH * (float)WW;
  float focal = acc[0] / N;
  float dice = 0.f;
  for (int i = 0; i < IMGS; ++i) {
    float inter = acc[1 + i], sg = acc[17 + i], tf = acc[33 + i];
    dice += (2.f * inter + 1e-6f) / (sg + tf + 1e-6f);
  }
  dice = 1.f - dice / (float)IMGS;
  float bnd = 0.f;
  for (int i = 0; i < IMGS; ++i) {
    float sp = acc[49 + 2 * i + 0], mp = acc[81 + 2 * i + 0];  // pos, (1-s)
    float sn = acc[49 + 2 * i + 1], mn = acc[81 + 2 * i + 1];  // neg, s
    bnd += sp / (mp + 1e-8f) + sn / (mn + 1e-8f);
  }
  bnd /= N;
  out[0] = 0.3f * focal + 0.4f * dice + 0.3f * bnd;
}

// ============================ launcher ============================

extern "C" void kernel_launch(void* const* d_in, const int* in_sizes, int n_in,
                              void* d_out, int out_size, void* d_ws,
                              size_t ws_size, hipStream_t stream) {
  const float* pred = (const float*)d_in[0];
  const int* target = (const int*)d_in[1];
  float* acc = (float*)d_ws;
  float* dfbase = acc + 128;  // 32 x (1024*1024) f32 Df buffers follow

  init_acc_kernel<<<1, 128, 0, stream>>>(acc);
  focal_dice_kernel<<<dim3(256, IMGS), 256, 0, stream>>>(pred, target, acc);
  edt_kernel<<<32, 1024, 0, stream>>>(target, pred, acc, dfbase);
  finalize_kernel<<<1, 32, 0, stream>>>(acc, (float*)d_out);
}